// Depth2Normals_22866405884569
// MI455X (gfx1250) — compile-verified
//
#include <hip/hip_runtime.h>
#include <cstdint>

namespace {
constexpr int Bn = 8;
constexpr int H  = 480;
constexpr int W  = 640;
constexpr int R  = 4;                 // int(2*SIGMA_D), SIGMA_D = 2.0
constexpr int BX = 16, BY = 16;
constexpr int TW  = BX + 2 * R;       // 24  (bilateral tile)
constexpr int THt = BY + 2 * R;       // 24
constexpr int MT  = 32;               // fused-median tile (halo 8)
constexpr int MH  = 8;                // median halo
constexpr int ITERS = 7;              // median-fill applications
constexpr float INV2SR = 50.0f;       // 1/(2*SIGMA_R^2), SIGMA_R = 0.1
constexpr float DMIN = 0.1f, DMAX = 6.0f;

__device__ __forceinline__ void async_load_f32(unsigned lds_addr,
                                               unsigned byte_off,
                                               const float* base) {
  asm volatile("global_load_async_to_lds_b32 %0, %1, %2"
               :
               : "v"(lds_addr), "v"(byte_off), "s"(base)
               : "memory");
}
__device__ __forceinline__ void wait_async0() {
  asm volatile("s_wait_asynccnt 0" ::: "memory");
}
}

// ---------------------------------------------------------------------------
// Kernel 0: clear the had_holes flag
// ---------------------------------------------------------------------------
__global__ void dn_init_flag(int* __restrict__ flag) {
  if (threadIdx.x == 0 && blockIdx.x == 0) *flag = 0;
}

// ---------------------------------------------------------------------------
// Kernel 1: 9x9 bilateral filter, async global->LDS halo staging.
// ---------------------------------------------------------------------------
__global__ void __launch_bounds__(256)
dn_bilateral(const float* __restrict__ src, float* __restrict__ dst,
             int* __restrict__ flag) {
  __shared__ float tile[THt * TW];

  const int tx  = threadIdx.x, ty = threadIdx.y;
  const int tid = ty * BX + tx;
  const int b   = blockIdx.z;
  const int gx0 = blockIdx.x * BX;
  const int gy0 = blockIdx.y * BY;

  for (int i = tid; i < THt * TW; i += BX * BY) tile[i] = 0.0f;
  __syncthreads();

  for (int i = tid; i < THt * TW; i += BX * BY) {
    const int ly = i / TW, lx = i % TW;
    const int gy = gy0 + ly - R;
    const int gx = gx0 + lx - R;
    if (gy >= 0 && gy < H && gx >= 0 && gx < W) {
      async_load_f32((unsigned)(uintptr_t)(&tile[i]),
                     (unsigned)((((b * H) + gy) * W + gx) * 4), src);
    }
  }
  wait_async0();
  __syncthreads();

  const int gx = gx0 + tx;
  const int gy = gy0 + ty;
  const float x = tile[(ty + R) * TW + (tx + R)];

  if (x == 0.0f) atomicOr(flag, 1);   // had_holes = any(d0 == 0)

  // exp(-k/8) for k = 0..32 (spatial Gaussian, sigma_d = 2)
  constexpr float WS[33] = {
    1.00000000f, 0.88249690f, 0.77880078f, 0.68728928f, 0.60653066f,
    0.53526143f, 0.47236655f, 0.41686207f, 0.36787944f, 0.32465247f,
    0.28650480f, 0.25283960f, 0.22313016f, 0.19691168f, 0.17377394f,
    0.15335497f, 0.13533528f, 0.11943297f, 0.10539922f, 0.09301449f,
    0.08208500f, 0.07243976f, 0.06392786f, 0.05641614f, 0.04978707f,
    0.04393693f, 0.03877421f, 0.03421812f, 0.03019738f, 0.02664910f,
    0.02351775f, 0.02075434f, 0.01831564f };

  float num = 0.0f, den = 0.0f;
  if (x > 0.0f) {
#pragma unroll
    for (int dy = -R; dy <= R; ++dy) {
#pragma unroll
      for (int dx = -R; dx <= R; ++dx) {
        const float nb = tile[(ty + R + dy) * TW + (tx + R + dx)];
        const float d  = x - nb;
        float w = WS[dy * dy + dx * dx] * __expf(-d * d * INV2SR);
        w = (nb > 0.0f) ? w : 0.0f;
        num += w * nb;
        den += w;
      }
    }
  }
  const float out = (x > 0.0f && den > 0.0f) ? (num / den) : 0.0f;
  dst[(size_t)b * H * W + (size_t)gy * W + gx] = out;
}

// ---------------------------------------------------------------------------
// Median-fill update for one cell from an LDS tile (zero = hole).
// Neighbors outside the image are 0 in the tile, matching jnp.pad.
// ---------------------------------------------------------------------------
__device__ __forceinline__ float median_update(const float* tile, int r,
                                               int c) {
  const float ctr = tile[r * MT + c];
  if (ctr != 0.0f) return ctr;
  float v[9];
  int k = 0;
#pragma unroll
  for (int dy = -1; dy <= 1; ++dy) {
#pragma unroll
    for (int dx = -1; dx <= 1; ++dx) v[k++] = tile[(r + dy) * MT + (c + dx)];
  }
  int cnt = 0;
#pragma unroll
  for (int i = 0; i < 9; ++i) cnt += (v[i] > 0.0f) ? 1 : 0;
  if (cnt == 0) return 0.0f;

  float sv[9];
#pragma unroll
  for (int i = 0; i < 9; ++i) sv[i] = (v[i] > 0.0f) ? v[i] : 1e10f;
#pragma unroll
  for (int i = 0; i < 8; ++i) {
#pragma unroll
    for (int j = 0; j < 8 - i; ++j) {
      const float lo = fminf(sv[j], sv[j + 1]);
      const float hi = fmaxf(sv[j], sv[j + 1]);
      sv[j] = lo; sv[j + 1] = hi;
    }
  }
  const int mi = (cnt - 1) >> 1;   // lower median index, 0..4
  float med = sv[0];
  med = (mi > 0) ? sv[1] : med;
  med = (mi > 1) ? sv[2] : med;
  med = (mi > 2) ? sv[3] : med;
  med = (mi > 3) ? sv[4] : med;
  return med;
}

// ---------------------------------------------------------------------------
// Kernel 2: all 7 median-fill passes fused in LDS.
// 32x32 tile (halo 8) staged with async global->LDS; 7 in-LDS iterations;
// central 16x16 written out. Margin induction guarantees exactness.
// ---------------------------------------------------------------------------
__global__ void __launch_bounds__(256)
dn_median_fused(const float* __restrict__ src, float* __restrict__ dst) {
  __shared__ float tile[MT * MT];

  const int tx  = threadIdx.x, ty = threadIdx.y;
  const int tid = ty * BX + tx;
  const int b   = blockIdx.z;
  const int gx0 = blockIdx.x * BX;
  const int gy0 = blockIdx.y * BY;

  // Zero-init (out-of-image cells stay 0 forever == zero padding).
#pragma unroll
  for (int j = 0; j < 4; ++j) tile[tid + j * 256] = 0.0f;
  __syncthreads();

  // Async stage the 32x32 region.
#pragma unroll
  for (int j = 0; j < 4; ++j) {
    const int i = tid + j * 256;
    const int r = i >> 5, c = i & 31;
    const int gy = gy0 - MH + r;
    const int gx = gx0 - MH + c;
    if (gy >= 0 && gy < H && gx >= 0 && gx < W) {
      async_load_f32((unsigned)(uintptr_t)(&tile[i]),
                     (unsigned)((((b * H) + gy) * W + gx) * 4), src);
    }
  }
  wait_async0();
  __syncthreads();

  // 7 median-fill iterations in LDS.
  for (int it = 0; it < ITERS; ++it) {
    float nv[4];
#pragma unroll
    for (int j = 0; j < 4; ++j) {
      const int i = tid + j * 256;
      const int r = i >> 5, c = i & 31;
      const int gy = gy0 - MH + r;
      const int gx = gx0 - MH + c;
      const bool in_img  = (gy >= 0) && (gy < H) && (gx >= 0) && (gx < W);
      const bool in_core = (r >= 1) && (r <= MT - 2) && (c >= 1) && (c <= MT - 2);
      nv[j] = (in_img && in_core) ? median_update(tile, r, c) : tile[i];
    }
    __syncthreads();
#pragma unroll
    for (int j = 0; j < 4; ++j) tile[tid + j * 256] = nv[j];
    __syncthreads();
  }

  dst[(size_t)b * H * W + (size_t)(gy0 + ty) * W + (gx0 + tx)] =
      tile[(ty + MH) * MT + (tx + MH)];
}

// ---------------------------------------------------------------------------
// Kernel 3: back-project + cross-product normals, zero border, (B,3,H,W) out.
// ---------------------------------------------------------------------------
__global__ void __launch_bounds__(256)
dn_normals(const float* __restrict__ d0, const float* __restrict__ dfill,
           const float* __restrict__ intr, const int* __restrict__ flag,
           float* __restrict__ out) {
  const int idx = blockIdx.x * 256 + threadIdx.x;
  if (idx >= Bn * H * W) return;
  const int x = idx % W;
  const int t = idx / W;
  const int y = t % H;
  const int b = t / H;

  const float* s = ((*flag) != 0 ? dfill : d0) + (size_t)b * H * W;
  const size_t ob = (size_t)b * 3 * H * W;

  float nx = 0.0f, ny = 0.0f, nz = 0.0f;
  if (x > 0 && x < W - 1 && y > 0 && y < H - 1) {
    const float fx = intr[b * 4 + 0];
    const float fy = intr[b * 4 + 1];
    const float cx = intr[b * 4 + 2];
    const float cy = intr[b * 4 + 3];

    const float dc = s[y * W + x];
    const float dl = s[y * W + (x - 1)];
    const float dr = s[y * W + (x + 1)];
    const float du = s[(y - 1) * W + x];
    const float dd = s[(y + 1) * W + x];

    auto cam = [&](float u, float v, float d, float& X, float& Y, float& Z,
                   bool& ok) {
      ok = (d >= DMIN) && (d <= DMAX);
      X = ok ? (u - cx) * d / fx : 0.0f;
      Y = ok ? (v - cy) * d / fy : 0.0f;
      Z = ok ? d : 0.0f;
    };

    float Lx, Ly, Lz, Rx, Ry, Rz, Ux, Uy, Uz, Dx, Dy, Dz;
    bool okL, okR, okU, okD;
    cam((float)(x - 1), (float)y, dl, Lx, Ly, Lz, okL);
    cam((float)(x + 1), (float)y, dr, Rx, Ry, Rz, okR);
    cam((float)x, (float)(y - 1), du, Ux, Uy, Uz, okU);
    cam((float)x, (float)(y + 1), dd, Dx, Dy, Dz, okD);
    const bool okC = (dc >= DMIN) && (dc <= DMAX);

    // dy_vec = camDown - camUp, dx_vec = camRight - camLeft
    const float ax = Dx - Ux, ay = Dy - Uy, az = Dz - Uz;
    const float bx = Rx - Lx, by = Ry - Ly, bz = Rz - Lz;
    // n = cross(dy_vec, dx_vec)
    nx = ay * bz - az * by;
    ny = az * bx - ax * bz;
    nz = ax * by - ay * bx;

    const bool z_ok = okL && okR && okU && okD && okC;
    const float nn = sqrtf(nx * nx + ny * ny + nz * nz);
    const bool good = (nn > 1e-8f) && z_ok;
    const float inv = good ? (1.0f / nn) : 0.0f;
    nx *= inv; ny *= inv; nz *= inv;
  }
  const size_t p = (size_t)y * W + x;
  out[ob + 0 * (size_t)H * W + p] = nx;
  out[ob + 1 * (size_t)H * W + p] = ny;
  out[ob + 2 * (size_t)H * W + p] = nz;
}

// ---------------------------------------------------------------------------
extern "C" void kernel_launch(void* const* d_in, const int* in_sizes, int n_in,
                              void* d_out, int out_size, void* d_ws,
                              size_t ws_size, hipStream_t stream) {
  (void)in_sizes; (void)n_in; (void)out_size; (void)ws_size;
  const float* depth = (const float*)d_in[0];      // (B,1,H,W) f32
  const float* intr  = (const float*)d_in[1];      // (B,4)     f32
  float* out = (float*)d_out;                      // (B,3,H,W) f32

  const size_t N = (size_t)Bn * H * W;             // 2,457,600
  float* P = out;                                  // scratch plane inside out
  float* Q = (float*)d_ws;                         // N floats of workspace
  int* flag = (int*)((char*)d_ws + N * sizeof(float));

  dn_init_flag<<<1, 1, 0, stream>>>(flag);

  dim3 bgrid(W / BX, H / BY, Bn);                  // 40 x 30 x 8
  dn_bilateral<<<bgrid, dim3(BX, BY), 0, stream>>>(depth, P, flag);

  // All 7 median-fill passes fused: P -> Q (Q does not alias out)
  dn_median_fused<<<bgrid, dim3(BX, BY), 0, stream>>>(P, Q);

  const int nblk = (int)(N / 256);                 // 9600, exact
  dn_normals<<<nblk, 256, 0, stream>>>(depth, Q, intr, flag, out);
}